// DeltaNetModel_84859963834847
// MI455X (gfx1250) — compile-verified
//
#include <hip/hip_runtime.h>
#include <hip/hip_bf16.h>

// ---------------- problem constants (match reference) ----------------
#define S_LEN  2048
#define B_DIM  8
#define HID    512
#define NHEAD  8
#define HDIM   64
#define FF_DIM 2048
#define PERH   193            // 3*D + 1
#define QKVB   1544           // H * PERH
#define MROWS  (S_LEN * B_DIM) // 16384 rows, [S,B] flattened: row = s*B + b

typedef float v2f __attribute__((ext_vector_type(2)));
typedef float v8f __attribute__((ext_vector_type(8)));

// =====================================================================
// Generic fp32 WMMA GEMM:  C[m,n] = sum_k A[m,k] * W[n,k] (+bias)(+res)(relu)
// Block tile 64(M) x 64(N), 128 threads = 4 wave32, K staged in LDS chunks of 16.
// TRANS_IN : A row m is read from x laid out [B,S,*]  (m = s*B+b -> b*S+s)
// TRANS_OUT: C row m is written to [B,S,*]            (m = s*B+b -> b*S+s)
// =====================================================================
template<bool TRANS_IN, bool TRANS_OUT, bool RELU, bool HAS_BIAS, bool HAS_RES>
__global__ __launch_bounds__(128) void gemm_wmma_kernel(
    const float* __restrict__ A, const float* __restrict__ W,
    const float* __restrict__ bias, const float* __restrict__ res,
    float* __restrict__ Cout, int N, int K)
{
    __shared__ float As[64][17];
    __shared__ float Bs[64][17];

    const int tid  = threadIdx.x;
    const int lane = tid & 31;
    const int wid  = tid >> 5;           // wave 0..3 -> M sub-tile
    const int lrow = lane & 15;
    const int khalf = (lane >> 4) * 2;   // f32 WMMA A/B K split across lane halves
    const int rhalf = (lane >> 4) * 8;   // C/D row split across lane halves

    const int n0 = blockIdx.x * 64;
    const int m0 = blockIdx.y * 64;

    // per-thread staging assignment: row = tid/2 (0..63), col half = (tid&1)*8
    const int ar = tid >> 1;
    const int ac = (tid & 1) * 8;

    long a_row = (long)m0 + ar;
    if (TRANS_IN) { long s = a_row / B_DIM; long b = a_row % B_DIM; a_row = b * (long)S_LEN + s; }
    const float* Arow = A + a_row * (long)K;

    const long w_row = (long)n0 + ar;
    const bool wvalid = (w_row < (long)N);
    const float* Wrow = W + (wvalid ? w_row * (long)K : 0);

    v8f acc[4] = {};

    for (int k0 = 0; k0 < K; k0 += 16) {
        float4 av0 = *(const float4*)(Arow + k0 + ac);
        float4 av1 = *(const float4*)(Arow + k0 + ac + 4);
        float4 wv0 = make_float4(0.f, 0.f, 0.f, 0.f), wv1 = wv0;
        if (wvalid) {
            wv0 = *(const float4*)(Wrow + k0 + ac);
            wv1 = *(const float4*)(Wrow + k0 + ac + 4);
        }
        As[ar][ac + 0] = av0.x; As[ar][ac + 1] = av0.y;
        As[ar][ac + 2] = av0.z; As[ar][ac + 3] = av0.w;
        As[ar][ac + 4] = av1.x; As[ar][ac + 5] = av1.y;
        As[ar][ac + 6] = av1.z; As[ar][ac + 7] = av1.w;
        Bs[ar][ac + 0] = wv0.x; Bs[ar][ac + 1] = wv0.y;
        Bs[ar][ac + 2] = wv0.z; Bs[ar][ac + 3] = wv0.w;
        Bs[ar][ac + 4] = wv1.x; Bs[ar][ac + 5] = wv1.y;
        Bs[ar][ac + 6] = wv1.z; Bs[ar][ac + 7] = wv1.w;
        __syncthreads();

        #pragma unroll
        for (int kk = 0; kk < 4; ++kk) {
            const int kb = kk * 4 + khalf;
            v2f a;
            a.x = As[(wid << 4) + lrow][kb];
            a.y = As[(wid << 4) + lrow][kb + 1];
            #pragma unroll
            for (int nt = 0; nt < 4; ++nt) {
                v2f b;
                b.x = Bs[(nt << 4) + lrow][kb];
                b.y = Bs[(nt << 4) + lrow][kb + 1];
                acc[nt] = __builtin_amdgcn_wmma_f32_16x16x4_f32(
                    false, a, false, b, (short)0, acc[nt], false, false);
            }
        }
        __syncthreads();
    }

    #pragma unroll
    for (int nt = 0; nt < 4; ++nt) {
        const long n = (long)n0 + (nt << 4) + lrow;
        if (n >= (long)N) continue;
        float bv = 0.f;
        if (HAS_BIAS) bv = bias[n];
        #pragma unroll
        for (int v = 0; v < 8; ++v) {
            const long m = (long)m0 + (wid << 4) + v + rhalf;
            float x = acc[nt][v] + bv;
            if (HAS_RES) x += res[m * (long)N + n];
            if (RELU) x = fmaxf(x, 0.f);
            long om = m;
            if (TRANS_OUT) { long s = m / B_DIM; long b = m % B_DIM; om = b * (long)S_LEN + s; }
            Cout[om * (long)N + n] = x;
        }
    }
}

// =====================================================================
// LayerNorm over rows of 512; one wave32 per row, 8 rows per block.
// =====================================================================
__global__ __launch_bounds__(256) void layernorm_kernel(
    const float* __restrict__ X, const float* __restrict__ G,
    const float* __restrict__ Bt, float* __restrict__ Y)
{
    const int lane = threadIdx.x & 31;
    const int wid  = threadIdx.x >> 5;
    const long row = (long)blockIdx.x * 8 + wid;
    const float* xr = X + row * (long)HID;

    float vals[16];
    float s = 0.f;
    #pragma unroll
    for (int c = 0; c < 4; ++c) {
        float4 v = *(const float4*)(xr + c * 128 + lane * 4);
        vals[c*4+0] = v.x; vals[c*4+1] = v.y; vals[c*4+2] = v.z; vals[c*4+3] = v.w;
        s += v.x + v.y + v.z + v.w;
    }
    #pragma unroll
    for (int o = 16; o > 0; o >>= 1) s += __shfl_xor(s, o, 32);
    const float mean = s * (1.f / HID);
    float vs = 0.f;
    #pragma unroll
    for (int i = 0; i < 16; ++i) { float d = vals[i] - mean; vs += d * d; }
    #pragma unroll
    for (int o = 16; o > 0; o >>= 1) vs += __shfl_xor(vs, o, 32);
    const float inv = rsqrtf(vs * (1.f / HID) + 1e-5f);

    float* yr = Y + row * (long)HID;
    #pragma unroll
    for (int c = 0; c < 4; ++c) {
        const int col = c * 128 + lane * 4;
        float4 g  = *(const float4*)(G + col);
        float4 bb = *(const float4*)(Bt + col);
        float4 o4;
        o4.x = (vals[c*4+0] - mean) * inv * g.x + bb.x;
        o4.y = (vals[c*4+1] - mean) * inv * g.y + bb.y;
        o4.z = (vals[c*4+2] - mean) * inv * g.z + bb.z;
        o4.w = (vals[c*4+3] - mean) * inv * g.w + bb.w;
        *(float4*)(yr + col) = o4;
    }
}

// =====================================================================
// Per-(row,head) activations on qkvb: softmax(q), softmax(k), sigmoid(beta).
// 64 threads (2 waves), LDS tree reductions.
// =====================================================================
__device__ __forceinline__ float red64_max(float v, float* red, int t) {
    red[t] = v; __syncthreads();
    #pragma unroll
    for (int s2 = 32; s2 > 0; s2 >>= 1) {
        if (t < s2) red[t] = fmaxf(red[t], red[t + s2]);
        __syncthreads();
    }
    float r = red[0]; __syncthreads();
    return r;
}
__device__ __forceinline__ float red64_sum(float v, float* red, int t) {
    red[t] = v; __syncthreads();
    #pragma unroll
    for (int s2 = 32; s2 > 0; s2 >>= 1) {
        if (t < s2) red[t] = red[t] + red[t + s2];
        __syncthreads();
    }
    float r = red[0]; __syncthreads();
    return r;
}

__global__ __launch_bounds__(64) void qkvb_act_kernel(float* __restrict__ Q)
{
    __shared__ float red[64];
    const long row = blockIdx.x;
    const int  h   = blockIdx.y;
    float* base = Q + row * (long)QKVB + (long)h * PERH;
    const int t = threadIdx.x;

    #pragma unroll
    for (int part = 0; part < 2; ++part) {          // 0: q, 1: k
        float* p = base + part * 64;
        float x  = p[t];
        float mx = red64_max(x, red, t);
        float e  = expf(x - mx);
        float sm = red64_sum(e, red, t);
        p[t] = e / sm;
    }
    if (t == 0) {
        float bb = base[192];
        base[192] = 1.f / (1.f + expf(-bb));        // sigmoid(beta)
    }
}

// =====================================================================
// Delta-rule scan. One block per (b,h): W (64x64) lives in registers,
// 16 elems/thread (row = tid/4, 16 cols). Quad reductions via shfl_xor.
// =====================================================================
__global__ __launch_bounds__(256) void delta_scan_kernel(
    const float* __restrict__ qkvb, float* __restrict__ attn)
{
    const int b  = blockIdx.x;
    const int h  = blockIdx.y;
    const int tid = threadIdx.x;
    const int r  = tid >> 2;        // row 0..63
    const int cg = tid & 3;         // col group
    const int cb = cg * 16;

    __shared__ float q_s[64], k_s[64], v_s[64];
    __shared__ float beta_s;

    float Wr[16];
    #pragma unroll
    for (int j = 0; j < 16; ++j) Wr[j] = 0.f;

    for (int t = 0; t < S_LEN; ++t) {
        const long roff = ((long)t * B_DIM + b) * (long)QKVB + (long)h * PERH;
        if (tid < 64) {
            q_s[tid] = qkvb[roff + tid];
            k_s[tid] = qkvb[roff + 64 + tid];
            v_s[tid] = qkvb[roff + 128 + tid];
            if (tid == 0) beta_s = qkvb[roff + 192];
        } else if (tid == 64 && (t + 1) < S_LEN) {
            __builtin_prefetch(qkvb + ((long)(t + 1) * B_DIM + b) * (long)QKVB + (long)h * PERH, 0, 1);
        }
        __syncthreads();

        float kv[16];
        float p = 0.f;
        #pragma unroll
        for (int j = 0; j < 16; ++j) { kv[j] = k_s[cb + j]; p += Wr[j] * kv[j]; }
        p += __shfl_xor(p, 1, 32);
        p += __shfl_xor(p, 2, 32);                  // wk[r] in all 4 quad lanes

        const float vn = beta_s * (v_s[r] - p);     // beta*(v - W k)

        float o = 0.f;
        #pragma unroll
        for (int j = 0; j < 16; ++j) { Wr[j] += vn * kv[j]; o += Wr[j] * q_s[cb + j]; }
        o += __shfl_xor(o, 1, 32);
        o += __shfl_xor(o, 2, 32);                  // out[r] = (W_new q)[r]

        if (cg == 0) attn[((long)t * B_DIM + b) * (long)HID + h * HDIM + r] = o;
        __syncthreads();
    }
}

// =====================================================================
// Host orchestration
// =====================================================================
extern "C" void kernel_launch(void* const* d_in, const int* in_sizes, int n_in,
                              void* d_out, int out_size, void* d_ws, size_t ws_size,
                              hipStream_t stream)
{
    const float* x         = (const float*)d_in[0];   // [B,S,IN]
    const float* ip_w      = (const float*)d_in[1];   // [HID,IN]
    const float* ip_b      = (const float*)d_in[2];   // [HID]
    const float* fw_ln_g   = (const float*)d_in[3];   // [2,HID]
    const float* fw_ln_b   = (const float*)d_in[4];   // [2,HID]
    const float* fw_slow_w = (const float*)d_in[5];   // [2,QKVB,HID]
    const float* fw_out_w  = (const float*)d_in[6];   // [2,HID,H*D]
    const float* ff_ln_g   = (const float*)d_in[7];   // [HID]
    const float* ff_ln_b   = (const float*)d_in[8];   // [HID]
    const float* ff_w1     = (const float*)d_in[9];   // [FF,HID]
    const float* ff_b1     = (const float*)d_in[10];  // [FF]
    const float* ff_w2     = (const float*)d_in[11];  // [HID,FF]
    const float* ff_b2     = (const float*)d_in[12];  // [HID]
    float* out = (float*)d_out;

    char* ws = (char*)d_ws;
    const size_t SZ_ACT = (size_t)MROWS * HID * sizeof(float);     // 33.5 MB
    float* act  = (float*)(ws);                                     // [S,B,HID]
    float* ln   = (float*)(ws + SZ_ACT);                            // [S,B,HID]
    float* attn = (float*)(ws + 2 * SZ_ACT);                        // [S,B,H*D]
    float* big  = (float*)(ws + 3 * SZ_ACT);                        // qkvb / ffn-hidden

    const dim3 blk(128);
    const int MT = MROWS / 64;   // 256 M-tiles

    // 1) input projection: [B,S,IN] x ip_w^T + ip_b  -> act [S,B,HID]
    gemm_wmma_kernel<true, false, false, true, false>
        <<<dim3(HID / 64, MT), blk, 0, stream>>>(x, ip_w, ip_b, nullptr, act, HID, HID);

    for (int pass = 0; pass < 2; ++pass) {
        // ---- fast_ff ----
        layernorm_kernel<<<MROWS / 8, 256, 0, stream>>>(
            act, fw_ln_g + pass * HID, fw_ln_b + pass * HID, ln);

        gemm_wmma_kernel<false, false, false, false, false>
            <<<dim3((QKVB + 63) / 64, MT), blk, 0, stream>>>(
                ln, fw_slow_w + (long)pass * QKVB * HID, nullptr, nullptr, big, QKVB, HID);

        qkvb_act_kernel<<<dim3(MROWS, NHEAD), 64, 0, stream>>>(big);

        delta_scan_kernel<<<dim3(B_DIM, NHEAD), 256, 0, stream>>>(big, attn);

        if (pass == 0) {
            // out-proj + residual, in place into act
            gemm_wmma_kernel<false, false, false, false, true>
                <<<dim3(HID / 64, MT), blk, 0, stream>>>(
                    attn, fw_out_w, nullptr, act, act, HID, HID);

            // ---- FFN ----
            layernorm_kernel<<<MROWS / 8, 256, 0, stream>>>(act, ff_ln_g, ff_ln_b, ln);
            gemm_wmma_kernel<false, false, true, true, false>
                <<<dim3(FF_DIM / 64, MT), blk, 0, stream>>>(
                    ln, ff_w1, ff_b1, nullptr, big, FF_DIM, HID);
            gemm_wmma_kernel<false, false, false, true, true>
                <<<dim3(HID / 64, MT), blk, 0, stream>>>(
                    big, ff_w2, ff_b2, act, act, HID, FF_DIM);
        } else {
            // final out-proj + residual, transposed store to [B,S,HID]
            gemm_wmma_kernel<false, true, false, false, true>
                <<<dim3(HID / 64, MT), blk, 0, stream>>>(
                    attn, fw_out_w + (long)HID * HID, nullptr, act, out, HID, HID);
        }
    }
}